// FrechetMeanClassifier_63642825392364
// MI455X (gfx1250) — compile-verified
//
#include <hip/hip_runtime.h>

// ---------------------------------------------------------------------------
// SPD Frechet-mean classifier, all matrix functions via Newton-Schulz GEMMs
// so everything runs on the CDNA5 fp32 matrix cores (v_wmma_f32_16x16x4_f32).
// One workgroup (8 wave32s) per sample; four 128x132-f32 matrix buffers in
// LDS (270 KB of the 320 KB/WGP CDNA5 budget); X^{-1/2} cached in registers
// (64 VGPRs/thread) across the class loop.
//
// v3 changes vs round 2:
//  * gemm_core<MODE>: elementwise maps fused into the WMMA accumulator store
//    (G = 1.5I-0.5*AB for Newton-Schulz, AB + c*I for log-Taylor Horner,
//    and a store-less Frobenius-reduce epilogue for the final log matrix).
//    Removes a full 16K-element LDS read+write pass per NS iteration.
//  * Coupled-NS iteration 0 specialized (Z0=I => Z1=G0=1.5I-0.5*Y0):
//    1 GEMM instead of 3 + identity pass per invocation.
// ---------------------------------------------------------------------------

#define LDSW      132              // padded row stride (floats) to spread banks
#define MATF      (128 * LDSW)     // floats per matrix buffer
#define NTHREADS  256              // 8 waves of 32
#define EPSF      1e-6f

#define SIGN_ITERS 24
#define INV_ITERS  16
#define SQRT_ITERS 12
#define LOG_LEVELS 6
#define TAYLOR_M   8

typedef __attribute__((ext_vector_type(2))) float v2f;
typedef __attribute__((ext_vector_type(8))) float v8f;

#define GM_PLAIN 0
#define GM_G     1   // D = 1.5 I - 0.5 (A*B)
#define GM_DIAG  2   // D = A*B + diagAdd * I
#define GM_FRO   3   // no store; return ||A*B||_F^2 (block-wide)

// ---------------------------------------------------------------------------
// 128x128x128 fp32 GEMM in LDS using V_WMMA_F32_16X16X4_F32, with a
// compile-time fused epilogue. Wave w computes output rows [16w, 16w+16);
// 8 tiles of 16x16 across columns. Fragment layouts per CDNA5 ISA 7.12.2.
// Force-inlined so LDS address space is provable at every call site.
// ---------------------------------------------------------------------------
template <int MODE>
__device__ __attribute__((always_inline)) inline
float gemm_core(float* D, const float* A, const float* B, float diagAdd) {
  __syncthreads();                       // operands produced by prior phase
  const int lane = threadIdx.x & 31;
  const int wave = threadIdx.x >> 5;     // 0..7
  const int half = lane >> 4;            // 0 or 1
  const int l15  = lane & 15;

  v8f acc[8] = {};
  const float* arow = A + (wave * 16 + l15) * LDSW;

#pragma unroll 4
  for (int k4 = 0; k4 < 32; ++k4) {
    const int kb = k4 * 4 + half * 2;    // A vgpr0: K=kb (halves 0/1 -> K 0/2)
    v2f a;
    a.x = arow[kb];
    a.y = arow[kb + 1];
    const float* bptr = B + kb * LDSW + l15;
    // preload all 8 B fragments first: one DS-load clause, then 8 WMMAs
    v2f bf[8];
#pragma unroll
    for (int tc = 0; tc < 8; ++tc) {
      bf[tc].x = bptr[tc * 16];          // B row kb,   col tc*16+l15
      bf[tc].y = bptr[LDSW + tc * 16];   // B row kb+1
    }
#pragma unroll
    for (int tc = 0; tc < 8; ++tc) {
      acc[tc] = __builtin_amdgcn_wmma_f32_16x16x4_f32(
          /*neg_a=*/false, a, /*neg_b=*/false, bf[tc],
          /*c_mod=*/(short)0, acc[tc], /*reuse_a=*/false, /*reuse_b=*/false);
    }
  }
  // C/D layout: vgpr i holds row (i + 8*half) of the tile, col = l15
  float fro = 0.f;
#pragma unroll
  for (int i = 0; i < 8; ++i) {
    const int r = wave * 16 + half * 8 + i;
#pragma unroll
    for (int tc = 0; tc < 8; ++tc) {
      const int c = tc * 16 + l15;
      float v = acc[tc][i];
      if (MODE == GM_G)    v = (r == c ? 1.5f : 0.f) - 0.5f * v;
      if (MODE == GM_DIAG) v = v + (r == c ? diagAdd : 0.f);
      if (MODE == GM_FRO)  fro += v * v;
      else                 D[r * LDSW + c] = v;
    }
  }
  if (MODE == GM_FRO) {
    __shared__ float red2[NTHREADS];
    red2[threadIdx.x] = fro;
    __syncthreads();
    for (int off = NTHREADS / 2; off > 0; off >>= 1) {
      if (threadIdx.x < off) red2[threadIdx.x] += red2[threadIdx.x + off];
      __syncthreads();
    }
    float r2 = red2[0];
    __syncthreads();
    return r2;
  }
  __syncthreads();
  return 0.f;
}

__device__ __attribute__((always_inline)) inline
void gemm128(float* D, const float* A, const float* B) {
  gemm_core<GM_PLAIN>(D, A, B, 0.f);
}
__device__ __attribute__((always_inline)) inline
void gemm128_g(float* D, const float* A, const float* B) {
  gemm_core<GM_G>(D, A, B, 0.f);
}
__device__ __attribute__((always_inline)) inline
void gemm128_diag(float* D, const float* A, const float* B, float d) {
  gemm_core<GM_DIAG>(D, A, B, d);
}
__device__ __attribute__((always_inline)) inline
float gemm128_fro(const float* A, const float* B) {
  return gemm_core<GM_FRO>(nullptr, A, B, 0.f);
}

// ---------------------------------------------------------------------------
// block-wide helpers (all end in a barrier)
// ---------------------------------------------------------------------------
__device__ float blockFro2(const float* A) {
  __shared__ float red[NTHREADS];
  float s = 0.f;
  for (int i = threadIdx.x; i < 128 * 128; i += NTHREADS) {
    float v = A[(i >> 7) * LDSW + (i & 127)];
    s += v * v;
  }
  red[threadIdx.x] = s;
  __syncthreads();
  for (int off = NTHREADS / 2; off > 0; off >>= 1) {
    if (threadIdx.x < off) red[threadIdx.x] += red[threadIdx.x + off];
    __syncthreads();
  }
  float r = red[0];
  __syncthreads();
  return r;
}

__device__ void ew_scale(float* D, const float* A, float s) {
  for (int i = threadIdx.x; i < 128 * 128; i += NTHREADS) {
    int idx = (i >> 7) * LDSW + (i & 127);
    D[idx] = A[idx] * s;
  }
  __syncthreads();
}

__device__ void ew_g_from(float* D, const float* A) {  // D = 1.5 I - 0.5 A
  for (int i = threadIdx.x; i < 128 * 128; i += NTHREADS) {
    int r = i >> 7, c = i & 127;
    D[r * LDSW + c] = (r == c ? 1.5f : 0.f) - 0.5f * A[r * LDSW + c];
  }
  __syncthreads();
}

__device__ void ew_setdiag(float* D, float s) {  // D = s * I
  for (int i = threadIdx.x; i < 128 * 128; i += NTHREADS) {
    int r = i >> 7, c = i & 127;
    D[r * LDSW + c] = (r == c) ? s : 0.f;
  }
  __syncthreads();
}

__device__ void ew_proj(float* A, const float* Babs) {  // A = (A+|A|)/2 + eps I
  for (int i = threadIdx.x; i < 128 * 128; i += NTHREADS) {
    int r = i >> 7, c = i & 127;
    int idx = r * LDSW + c;
    A[idx] = 0.5f * (A[idx] + Babs[idx]) + (r == c ? EPSF : 0.f);
  }
  __syncthreads();
}

__device__ void ew_subI(float* A) {  // A <- A - I
  for (int i = threadIdx.x; i < 128 * 128; i += NTHREADS) {
    int r = i >> 7, c = i & 127;
    if (r == c) A[r * LDSW + c] -= 1.f;
  }
  __syncthreads();
}

__device__ inline float coefT(int j) { return ((j & 1) ? 1.f : -1.f) / (float)j; }

// ---------------------------------------------------------------------------
// SPD projection in place: bA (symmetric) -> (bA + |bA|)/2 + eps I.
// |A| = A * sign(A), sign via Newton-Schulz: S <- S * (1.5 I - 0.5 S^2).
// ---------------------------------------------------------------------------
__device__ void spd_project(float* bA, float* bS, float* bT, float* bO) {
  float f2 = blockFro2(bA);
  ew_scale(bS, bA, rsqrtf(f2 + 1e-30f));       // S0 = A/||A||_F  (||S||_2<=1)
  float* pS = bS;
  float* pO = bO;
#pragma clang loop unroll(disable)
  for (int it = 0; it < SIGN_ITERS; ++it) {
    gemm128_g(bT, pS, pS);                     // G = 1.5I - 0.5 S^2 (fused)
    gemm128(pO, pS, bT);                       // S <- S G
    float* t = pS; pS = pO; pO = t;
  }
  gemm128(bT, bA, pS);                         // |A| = A sign(A)
  ew_proj(bA, bT);
}

// ---------------------------------------------------------------------------
// Coupled Newton-Schulz: Y->sqrt(A/c), Z->(A/c)^{-1/2}.  c = ||A||_F.
// Iter 0 specialized: Z0 = I  =>  G0 = 1.5I - 0.5 Y0,  Y1 = Y0 G0,  Z1 = G0.
// ---------------------------------------------------------------------------
__device__ float* ns_invsqrt(float* bA, float* b1, float* b2, float* b3) {
  float f2 = blockFro2(bA);
  ew_scale(bA, bA, rsqrtf(f2 + 1e-30f));       // Y0 = A/c
  ew_g_from(b1, bA);                           // G0 (= Z1)
  gemm128(b2, bA, b1);                         // Y1 = Y0 G0
  float *pY = b2, *pZ = b1, *pT = bA, *pO = b3;
#pragma clang loop unroll(disable)
  for (int it = 1; it < INV_ITERS; ++it) {
    gemm128_g(pT, pZ, pY);                     // G = 1.5I - 0.5 Z Y (fused)
    gemm128(pO, pY, pT);                       // Ynew = Y G
    gemm128(pY, pT, pZ);                       // Znew = G Z  (into old Y)
    float* nY = pO; float* nZ = pY; float* nT = pZ; float* nO = pT;
    pY = nY; pZ = nZ; pT = nT; pO = nO;
  }
  ew_scale(pZ, pZ, rsqrtf(sqrtf(f2 + 1e-30f)));  // A^{-1/2} = Z * c^{-1/2}
  return pZ;
}

__device__ float* ns_sqrt(float* bA, float** f1, float** f2, float** f3) {
  float f2n = blockFro2(bA);
  ew_scale(bA, bA, rsqrtf(f2n + 1e-30f));      // Y0
  ew_g_from(*f1, bA);                          // G0 (= Z1)
  gemm128(*f2, bA, *f1);                       // Y1
  float *pY = *f2, *pZ = *f1, *pT = bA, *pO = *f3;
#pragma clang loop unroll(disable)
  for (int it = 1; it < SQRT_ITERS; ++it) {
    gemm128_g(pT, pZ, pY);
    gemm128(pO, pY, pT);
    gemm128(pY, pT, pZ);
    float* nY = pO; float* nZ = pY; float* nT = pZ; float* nO = pT;
    pY = nY; pZ = nZ; pT = nT; pO = nO;
  }
  ew_scale(pY, pY, sqrtf(sqrtf(f2n + 1e-30f)));  // sqrt(A) = Y * c^{1/2}
  *f1 = pZ; *f2 = pT; *f3 = pO;
  return pY;
}

// ---------------------------------------------------------------------------
// Kernel 1: project the C prototypes onto SPD cone, store dense to d_ws.
// ---------------------------------------------------------------------------
__global__ void __launch_bounds__(NTHREADS)
proj_proto_kernel(const float* __restrict__ Pr, float* __restrict__ Pp) {
  extern __shared__ float smem[];
  float* b0 = smem;
  float* b1 = smem + MATF;
  float* b2 = smem + 2 * MATF;
  float* b3 = smem + 3 * MATF;
  const float* Ac = Pr + (size_t)blockIdx.x * 128 * 128;
  for (int i = threadIdx.x; i < 128 * 128; i += NTHREADS) {
    int r = i >> 7, c = i & 127;
    b0[r * LDSW + c] = 0.5f * (Ac[r * 128 + c] + Ac[c * 128 + r]);
  }
  __syncthreads();
  spd_project(b0, b1, b2, b3);
  float* Dst = Pp + (size_t)blockIdx.x * 128 * 128;
  for (int i = threadIdx.x; i < 128 * 128; i += NTHREADS) {
    int r = i >> 7, c = i & 127;
    Dst[i] = b0[r * LDSW + c];
  }
}

// ---------------------------------------------------------------------------
// Kernel 2: per sample b: project X_b, form Xis = Xp^{-1/2}; per class:
// M = Xis P Xis, dist = 2^s * || log-Taylor( M^{1/2^s} ) ||_F, out = -dist.
// ---------------------------------------------------------------------------
__global__ void __launch_bounds__(NTHREADS)
spd_dist_kernel(const float* __restrict__ X, const float* __restrict__ Pp,
                float* __restrict__ out, int C) {
  extern __shared__ float smem[];
  float* b0 = smem;
  float* b1 = smem + MATF;
  float* b2 = smem + 2 * MATF;
  float* b3 = smem + 3 * MATF;
  const int bid = blockIdx.x;
  const float* Xb = X + (size_t)bid * 128 * 128;

  // symmetrize into LDS
  for (int i = threadIdx.x; i < 128 * 128; i += NTHREADS) {
    int r = i >> 7, c = i & 127;
    b0[r * LDSW + c] = 0.5f * (Xb[r * 128 + c] + Xb[c * 128 + r]);
  }
  __syncthreads();

  spd_project(b0, b1, b2, b3);                 // b0 = Xp
  float* pZ = ns_invsqrt(b0, b1, b2, b3);      // pZ = Xp^{-1/2}

  // park Xis in registers (64 VGPRs/thread) to free all 4 LDS buffers
  float xis[64];
#pragma unroll
  for (int e = 0; e < 64; ++e) {
    int i = e * NTHREADS + threadIdx.x;
    xis[e] = pZ[(i >> 7) * LDSW + (i & 127)];
  }
  __syncthreads();

#pragma clang loop unroll(disable)
  for (int cls = 0; cls < C; ++cls) {
    // Xis -> b0
#pragma unroll
    for (int e = 0; e < 64; ++e) {
      int i = e * NTHREADS + threadIdx.x;
      b0[(i >> 7) * LDSW + (i & 127)] = xis[e];
    }
    // prototype -> b1
    const float* P = Pp + (size_t)cls * 128 * 128;
    for (int i = threadIdx.x; i < 128 * 128; i += NTHREADS) {
      b1[(i >> 7) * LDSW + (i & 127)] = P[i];
    }
    __syncthreads();

    gemm128(b2, b0, b1);                       // T1 = Xis * Pp
    gemm128(b1, b2, b0);                       // M  = T1 * Xis

    // prefetch next prototype into cache while the log-chain runs
    if (cls + 1 < C) {
      const float* Pn = Pp + (size_t)(cls + 1) * 128 * 128;
      __builtin_prefetch(Pn + (size_t)threadIdx.x * 64, 0, 1);
    }

    float* pA = b1;
    float* f1 = b0; float* f2 = b2; float* f3 = b3;
#pragma clang loop unroll(disable)
    for (int lvl = 0; lvl < LOG_LEVELS; ++lvl)
      pA = ns_sqrt(pA, &f1, &f2, &f3);         // pA <- sqrt(pA)

    ew_subI(pA);                               // E = A_s - I
    // Horner: q = c_m I; q = E q + c_j I (fused diag store, pointer swap)
    float* acc = f1; float* tmp = f2;
    ew_setdiag(acc, coefT(TAYLOR_M));
#pragma clang loop unroll(disable)
    for (int j = TAYLOR_M - 1; j >= 1; --j) {
      gemm128_diag(tmp, pA, acc, coefT(j));
      float* t = acc; acc = tmp; tmp = t;
    }
    // L = E q(E): store-less GEMM, Frobenius-reduce from accumulators
    float l2 = gemm128_fro(pA, acc);

    if (threadIdx.x == 0) {
      float dist = sqrtf(l2) * (float)(1 << LOG_LEVELS);  // ||log M||_F
      out[bid * C + cls] = -dist;
    }
    __syncthreads();
  }
}

// ---------------------------------------------------------------------------
extern "C" void kernel_launch(void* const* d_in, const int* in_sizes, int n_in,
                              void* d_out, int out_size, void* d_ws, size_t ws_size,
                              hipStream_t stream) {
  (void)n_in; (void)out_size; (void)ws_size;
  const float* X  = (const float*)d_in[0];     // [B,128,128] f32
  const float* Pr = (const float*)d_in[1];     // [C,128,128] f32
  float* out = (float*)d_out;                  // [B,C] f32
  float* Pp  = (float*)d_ws;                   // C*128*128 f32 scratch

  const int B = in_sizes[0] / (128 * 128);
  const int C = in_sizes[1] / (128 * 128);
  const size_t shmem = (size_t)4 * MATF * sizeof(float);  // 270336 B < 320KB/WGP

  hipFuncSetAttribute(reinterpret_cast<const void*>(proj_proto_kernel),
                      hipFuncAttributeMaxDynamicSharedMemorySize, (int)shmem);
  hipFuncSetAttribute(reinterpret_cast<const void*>(spd_dist_kernel),
                      hipFuncAttributeMaxDynamicSharedMemorySize, (int)shmem);

  proj_proto_kernel<<<C, NTHREADS, shmem, stream>>>(Pr, Pp);
  spd_dist_kernel<<<B, NTHREADS, shmem, stream>>>(X, Pp, out, C);
}